// DualDescriptorTS_9990093930563
// MI455X (gfx1250) — compile-verified
//
#include <hip/hip_runtime.h>

typedef __attribute__((ext_vector_type(2))) float v2f;
typedef __attribute__((ext_vector_type(8))) float v8f;

#define M_DIM   64
#define O_DIM   16
#define QTOT    1024          // M*O q-values per i
#define B_PER_WG 128
#define THREADS 256
#define XPITCH  (M_DIM + 1)   // pad to avoid 16-way LDS bank conflicts

__global__ __launch_bounds__(THREADS) void dd_ts_kernel(
    const float* __restrict__ k_tensor,   // [B]
    const int*   __restrict__ tok,        // [B]
    const float* __restrict__ emb,        // [VOCAB, 64]
    const float* __restrict__ P,          // [64, 64, 16] flat
    float*       __restrict__ out)        // [B, 64]
{
    __shared__ float s_invp[QTOT];
    __shared__ float s_P[QTOT];
    __shared__ float s_x[B_PER_WG * XPITCH];

    const int i      = blockIdx.x;               // output feature / period block
    const int b_base = blockIdx.y * B_PER_WG;
    const int tid    = threadIdx.x;

    // Stage 1/p and this i's slice of P (contiguous: P[i*1024 + q]).
    for (int q = tid; q < QTOT; q += THREADS) {
        s_P[q]    = P[i * QTOT + q];
        s_invp[q] = 1.0f / (float)(i * QTOT + q + 2);
    }
    // Stage x = emb[tok]: 128 rows x 64 cols, two threads per row.
    {
        const int row = tid >> 1;
        const int c0  = (tid & 1) * 32;
        const int t   = tok[b_base + row];
        const float* er = emb + t * M_DIM;
        float* xr = &s_x[row * XPITCH];
        #pragma unroll
        for (int c = 0; c < 32; ++c) xr[c0 + c] = er[c0 + c];
    }
    __syncthreads();

    const int lane  = tid & 31;
    const int wave  = tid >> 5;
    const int m     = lane & 15;   // A-matrix row (b offset in tile)
    const int kz    = lane >> 4;   // selects K-pair {0,1} vs {2,3}
    const int b_loc = wave * 16 + m;

    const float kval = k_tensor[b_base + b_loc];

    v8f acc = {};
    for (int j = 0; j < M_DIM; ++j) {
        const float xv = s_x[b_loc * XPITCH + j];
        const int   qj = j * O_DIM;
        #pragma unroll
        for (int t4 = 0; t4 < 4; ++t4) {
            const int q = qj + t4 * 4 + kz * 2;     // this lane's first K of the chunk
            const float ip0 = s_invp[q];
            const float ip1 = s_invp[q + 1];
            // v_cos_f32 input is in revolutions: cos(2*pi*k/p) == v_cos(k * (1/p)).
            float a0 = kval * ip0; a0 -= floorf(a0);
            float a1 = kval * ip1; a1 -= floorf(a1);
            const float c0 = __builtin_amdgcn_cosf(a0);
            const float c1 = __builtin_amdgcn_cosf(a1);
            v2f av, bv;
            av.x = xv * c0;  av.y = xv * c1;        // A[m, k] = x * phi
            bv.x = s_P[q];   bv.y = s_P[q + 1];     // B[k, n] = P (bcast over n)
            acc = __builtin_amdgcn_wmma_f32_16x16x4_f32(
                false, av, false, bv, (short)0, acc, false, false);
        }
    }

    // Every column of D is identical; column N=0 lives in lanes 0 (rows 0-7)
    // and 16 (rows 8-15). Out layout is [B, 64] row-major.
    if (m == 0) {
        const int b0 = b_base + wave * 16 + kz * 8;
        #pragma unroll
        for (int r = 0; r < 8; ++r)
            out[(b0 + r) * M_DIM + i] = acc[r];
    }
}

extern "C" void kernel_launch(void* const* d_in, const int* in_sizes, int n_in,
                              void* d_out, int out_size, void* d_ws, size_t ws_size,
                              hipStream_t stream) {
    const float* k_tensor = (const float*)d_in[0];   // [4096]
    const int*   tok      = (const int*)d_in[1];     // [4096]
    const float* emb      = (const float*)d_in[2];   // [256*64]
    const float* P        = (const float*)d_in[3];   // [64*64*16]
    float*       out      = (float*)d_out;           // [4096*64]

    const int B = in_sizes[0];                       // 4096
    dim3 grid(M_DIM, B / B_PER_WG);                  // (64, 32)
    dd_ts_kernel<<<grid, THREADS, 0, stream>>>(k_tensor, tok, emb, P, out);
}